// MinLinear_64433099374741
// MI455X (gfx1250) — compile-verified
//
#include <hip/hip_runtime.h>

// Tropical (min-plus) matmul: out[b,o] = min_i (W[o,i] + x[b,i])
// x: [B, IN] f32, W: [OUT, IN] f32 -> out: [B, OUT] f32
//
// VALU-bound problem (min-plus is not expressible with WMMA), so we target the
// CDNA5 async global->LDS path (ASYNCcnt) + wave32 register blocking instead.

constexpr int B_DIM   = 512;
constexpr int OUT_DIM = 1024;
constexpr int IN_DIM  = 1024;

constexpr int BM  = 64;        // x / out rows per block
constexpr int BN  = 64;        // W rows (out cols) per block
constexpr int BK  = 32;        // k tile
constexpr int LDK = BK + 4;    // padded LDS row stride (floats): 144B, 16B aligned,
                               // 4-row stride = 144 dwords -> bank offset 16 (no conflicts)
constexpr int NTHREADS = 256;  // 8 waves (wave32)

#if defined(__HIP_DEVICE_COMPILE__) && defined(__gfx1250__) && \
    __has_builtin(__builtin_amdgcn_global_load_async_to_lds_b128) && \
    __has_builtin(__builtin_amdgcn_s_wait_asynccnt)
#define USE_ASYNC 1
#else
#define USE_ASYNC 0
#endif

#if USE_ASYNC
typedef int v4i __attribute__((ext_vector_type(4)));
typedef __attribute__((address_space(1))) v4i* gv4_t;   // global-AS vector ptr
typedef __attribute__((address_space(3))) v4i* lv4_t;   // LDS-AS vector ptr
__device__ __forceinline__ void async_copy_b128(const float* g, float* l) {
  __builtin_amdgcn_global_load_async_to_lds_b128((gv4_t)g, (lv4_t)l, 0, 0);
}
#endif

__global__ __launch_bounds__(NTHREADS)
void minplus_matmul_kernel(const float* __restrict__ x,
                           const float* __restrict__ W,
                           float* __restrict__ out) {
  // [buffer][0 = x tile, 1 = W tile][row * LDK + k]
  __shared__ float lds[2][2][BM * LDK];

  const int tid   = threadIdx.x;
  const int bTile = blockIdx.y;   // over B
  const int oTile = blockIdx.x;   // over OUT

  // ---- loader mapping: 256 threads cover 64 rows x 2 b128-chunks (x and W) ----
  const int ldRow = tid >> 2;          // 0..63
  const int ldK   = (tid & 3) << 2;    // 0,4,8,12 (floats)

  const float* gx = x + (size_t)(bTile * BM + ldRow) * IN_DIM + ldK;
  const float* gw = W + (size_t)(oTile * BN + ldRow) * IN_DIM + ldK;

  auto issue = [&](int buf, int k0) {
    float* lx = &lds[buf][0][ldRow * LDK + ldK];
    float* lw = &lds[buf][1][ldRow * LDK + ldK];
#if USE_ASYNC
    async_copy_b128(gx + k0,      lx);
    async_copy_b128(gx + k0 + 16, lx + 16);
    async_copy_b128(gw + k0,      lw);
    async_copy_b128(gw + k0 + 16, lw + 16);
#else
    *(float4*)lx        = *(const float4*)(gx + k0);
    *(float4*)(lx + 16) = *(const float4*)(gx + k0 + 16);
    *(float4*)lw        = *(const float4*)(gw + k0);
    *(float4*)(lw + 16) = *(const float4*)(gw + k0 + 16);
#endif
  };

  // ---- compute mapping: 16x16 threads, 4x4 outputs each ----
  const int tx = tid & 15;   // column group (W rows)
  const int ty = tid >> 4;   // row group (x rows)

  float acc[4][4];
#pragma unroll
  for (int r = 0; r < 4; ++r)
#pragma unroll
    for (int c = 0; c < 4; ++c) acc[r][c] = __builtin_inff();

  constexpr int NKT = IN_DIM / BK;   // 32 k-tiles

  issue(0, 0);   // prologue: tile 0 into buffer 0

  for (int t = 0; t < NKT; ++t) {
    const int cur = t & 1;
    if (t + 1 < NKT) issue(cur ^ 1, (t + 1) * BK);
#if USE_ASYNC
    // Async loads complete in order: waiting for <=4 outstanding guarantees the
    // current tile's 4 copies have landed; barrier extends that to all waves.
    if (t + 1 < NKT) __builtin_amdgcn_s_wait_asynccnt(4);
    else             __builtin_amdgcn_s_wait_asynccnt(0);
#endif
    __syncthreads();

    const float* xrow = &lds[cur][0][(ty * 4) * LDK];
    const float* wrow = &lds[cur][1][(tx * 4) * LDK];

#pragma unroll
    for (int kc = 0; kc < BK; kc += 4) {
      float4 xa[4], wv[4];
#pragma unroll
      for (int r = 0; r < 4; ++r) xa[r] = *(const float4*)(xrow + r * LDK + kc);
#pragma unroll
      for (int c = 0; c < 4; ++c) wv[c] = *(const float4*)(wrow + c * LDK + kc);
#pragma unroll
      for (int r = 0; r < 4; ++r) {
#pragma unroll
        for (int c = 0; c < 4; ++c) {
          const float s0 = xa[r].x + wv[c].x;
          const float s1 = xa[r].y + wv[c].y;
          const float s2 = xa[r].z + wv[c].z;
          const float s3 = xa[r].w + wv[c].w;
          // linear fmin chain -> backend can fuse into v_min3_num_f32 pairs
          acc[r][c] = fminf(fminf(fminf(fminf(acc[r][c], s0), s1), s2), s3);
        }
      }
    }
    __syncthreads();   // protects buffer cur before it is refilled at t+2
  }

  // ---- epilogue: coalesced float4 stores ----
#pragma unroll
  for (int r = 0; r < 4; ++r) {
    float4 v = make_float4(acc[r][0], acc[r][1], acc[r][2], acc[r][3]);
    float* o = out + (size_t)(bTile * BM + ty * 4 + r) * OUT_DIM
                   + (size_t)(oTile * BN + tx * 4);
    *(float4*)o = v;
  }
}

extern "C" void kernel_launch(void* const* d_in, const int* in_sizes, int n_in,
                              void* d_out, int out_size, void* d_ws, size_t ws_size,
                              hipStream_t stream) {
  const float* x = (const float*)d_in[0];   // [512, 1024]
  const float* W = (const float*)d_in[1];   // [1024, 1024]
  float*     out = (float*)d_out;           // [512, 1024]
  (void)in_sizes; (void)n_in; (void)out_size; (void)d_ws; (void)ws_size;

  dim3 grid(OUT_DIM / BN, B_DIM / BM);      // (16, 8) = 128 workgroups
  minplus_matmul_kernel<<<grid, NTHREADS, 0, stream>>>(x, W, out);
}